// PIPPack_67070209294575
// MI455X (gfx1250) — compile-verified
//
#include <hip/hip_runtime.h>
#include <hip/hip_bf16.h>
#include <math.h>

typedef __attribute__((ext_vector_type(2))) float v2f;
typedef __attribute__((ext_vector_type(8))) float v8f;

#define BB 8
#define NN 2048
#define KNB 30
#define DD 128
#define EDIM 128
#define HH 4
#define CC 16
#define PQN 4
#define PVN 8
#define NODES (BB * NN)
#define PROJ_N 384
#define OUT_IN 704

// ---------------------------------------------------------------------------
// Kernel A: fused projection GEMM  (16384 x 128) @ (128 x [64|128|48|144])
// One wave per 16x16 output tile, V_WMMA_F32_16X16X4_F32, K-loop of 32.
// ---------------------------------------------------------------------------
__global__ void __launch_bounds__(128)
proj_gemm(const float* __restrict__ hV,
          const float* __restrict__ Wq,  const float* __restrict__ bq,
          const float* __restrict__ Wkv, const float* __restrict__ bkv,
          const float* __restrict__ Wqp, const float* __restrict__ bqp,
          const float* __restrict__ Wkvp,const float* __restrict__ bkvp,
          float* __restrict__ proj) {
  const int wave = threadIdx.x >> 5;
  const int lane = threadIdx.x & 31;
  const int row0 = blockIdx.x * 16;
  const int ntile = blockIdx.y * 4 + wave;       // 0..23
  const int n = ntile * 16 + (lane & 15);        // virtual output column 0..383
  const float* Wp; const float* bp; int ncols, col;
  if (n < 64)       { Wp = Wq;   bp = bq;   ncols = 64;  col = n; }
  else if (n < 192) { Wp = Wkv;  bp = bkv;  ncols = 128; col = n - 64; }
  else if (n < 240) { Wp = Wqp;  bp = bqp;  ncols = 48;  col = n - 192; }
  else              { Wp = Wkvp; bp = bkvp; ncols = 144; col = n - 240; }
  const int mrow = row0 + (lane & 15);
  const int koff = (lane < 16) ? 0 : 2;
  const float* arow = hV + (size_t)mrow * DD;
  v8f acc = {0.f, 0.f, 0.f, 0.f, 0.f, 0.f, 0.f, 0.f};
  for (int k0 = 0; k0 < DD; k0 += 4) {
    v2f av; av[0] = arow[k0 + koff];               av[1] = arow[k0 + koff + 1];
    v2f bv; bv[0] = Wp[(k0 + koff) * ncols + col]; bv[1] = Wp[(k0 + koff + 1) * ncols + col];
    acc = __builtin_amdgcn_wmma_f32_16x16x4_f32(false, av, false, bv, (short)0, acc, false, false);
  }
  const float bias = bp[col];
  const int mbase = row0 + ((lane < 16) ? 0 : 8);
#pragma unroll
  for (int r = 0; r < 8; ++r)
    proj[(size_t)(mbase + r) * PROJ_N + n] = acc[r] + bias;
}

// ---------------------------------------------------------------------------
// Kernel B: backbone frames + point rotation (one thread per node)
// ---------------------------------------------------------------------------
__global__ void frames_rotate(const float* __restrict__ X, const float* __restrict__ proj,
                              float* __restrict__ Rt, float* __restrict__ pts) {
  const int node = blockIdx.x * blockDim.x + threadIdx.x;
  if (node >= NODES) return;
  const float* xp = X + (size_t)node * 12;
  const float inv = 0.1f;  // 1/POS_SCALE
  const float nx = xp[0]*inv, ny = xp[1]*inv, nz = xp[2]*inv;
  const float cax = xp[3]*inv, cay = xp[4]*inv, caz = xp[5]*inv;
  const float cx = xp[6]*inv, cy = xp[7]*inv, cz = xp[8]*inv;
  const float v1x = cx - cax, v1y = cy - cay, v1z = cz - caz;
  const float v2x = nx - cax, v2y = ny - cay, v2z = nz - caz;
  const float r1 = rsqrtf(v1x*v1x + v1y*v1y + v1z*v1z + 1e-12f);
  const float e1x = v1x*r1, e1y = v1y*r1, e1z = v1z*r1;
  const float dd = e1x*v2x + e1y*v2y + e1z*v2z;
  const float u2x = v2x - e1x*dd, u2y = v2y - e1y*dd, u2z = v2z - e1z*dd;
  const float r2 = rsqrtf(u2x*u2x + u2y*u2y + u2z*u2z + 1e-12f);
  const float e2x = u2x*r2, e2y = u2y*r2, e2z = u2z*r2;
  const float e3x = e1y*e2z - e1z*e2y;
  const float e3y = e1z*e2x - e1x*e2z;
  const float e3z = e1x*e2y - e1y*e2x;
  // R row-major R[i*3+j]; columns are e1,e2,e3
  float R[9] = { e1x, e2x, e3x,  e1y, e2y, e3y,  e1z, e2z, e3z };
  float t0 = cax, t1 = cay, t2 = caz;
  float* rt = Rt + (size_t)node * 12;
#pragma unroll
  for (int i = 0; i < 9; ++i) rt[i] = R[i];
  rt[9] = t0; rt[10] = t1; rt[11] = t2;
  const float* pr = proj + (size_t)node * PROJ_N;
  float* po = pts + (size_t)node * 192;
  // q_pts: raw col = 192 + i*16 + m  ->  po[m*3 + i]  (m = h*PQ + p)
  for (int m = 0; m < 16; ++m) {
    const float px = pr[192 + m], py = pr[192 + 16 + m], pz = pr[192 + 32 + m];
    po[m*3 + 0] = R[0]*px + R[1]*py + R[2]*pz + t0;
    po[m*3 + 1] = R[3]*px + R[4]*py + R[5]*pz + t1;
    po[m*3 + 2] = R[6]*px + R[7]*py + R[8]*pz + t2;
  }
  // kv_pts: raw col = 240 + i*48 + m  (m = h*12 + pp)
  for (int m = 0; m < 48; ++m) {
    const float px = pr[240 + m], py = pr[240 + 48 + m], pz = pr[240 + 96 + m];
    const float ox = R[0]*px + R[1]*py + R[2]*pz + t0;
    const float oy = R[3]*px + R[4]*py + R[5]*pz + t1;
    const float oz = R[6]*px + R[7]*py + R[8]*pz + t2;
    const int h = m / 12, pp = m % 12;
    float* dst = (pp < PQN) ? (po + 48 + (h*PQN + pp) * 3)
                            : (po + 96 + (h*PVN + pp - PQN) * 3);
    dst[0] = ox; dst[1] = oy; dst[2] = oz;
  }
}

// ---------------------------------------------------------------------------
// Kernel C: fused attention per node. h_E staged once in LDS (single HBM pass,
// b128 loads), bias -> logits -> softmax -> o / o_pt / o_pair (WMMA).
// ---------------------------------------------------------------------------
__global__ void __launch_bounds__(128)
attn_kernel(const float* __restrict__ hE_g, const int* __restrict__ Eidx,
            const float* __restrict__ mask, const float* __restrict__ Wb,
            const float* __restrict__ bb_g, const float* __restrict__ hwv,
            const float* __restrict__ proj, const float* __restrict__ Rt,
            const float* __restrict__ pts, float* __restrict__ s_in) {
  const int node = blockIdx.x;       // b*N + i
  const int b = node / NN;
  const int tid = threadIdx.x;

  __shared__ __align__(16) float hE[32][EDIM];   // rows 30,31 zero-padded (WMMA B)
  __shared__ __align__(16) float kbn[KNB][64];
  __shared__ __align__(16) float vbn[KNB][64];
  __shared__ __align__(16) float kpt[KNB][48];
  __shared__ __align__(16) float vpt[KNB][96];
  __shared__ __align__(16) float apad[16][32];   // softmax weights, zero-padded (WMMA A)
  __shared__ float lg[KNB][HH];
  __shared__ float qv[64];
  __shared__ float qp[48];
  __shared__ float Rs[12];
  __shared__ int   Es[KNB];
  __shared__ float osum[64];
  __shared__ float optv[96];

  // stage h_E once: 30*128 floats = 960 b128 loads
  const float4* he4 = (const float4*)(hE_g + (size_t)node * KNB * EDIM);
  for (int e = tid; e < KNB * (EDIM / 4); e += 128) {
    const float4 v = he4[e];
    const int j = e >> 5;            // 32 float4 per row
    const int c = (e & 31) * 4;
    hE[j][c] = v.x; hE[j][c+1] = v.y; hE[j][c+2] = v.z; hE[j][c+3] = v.w;
  }
  hE[30][tid] = 0.f; hE[31][tid] = 0.f;
  for (int e = tid; e < 16 * 32; e += 128) apad[e >> 5][e & 31] = 0.f;
  if (tid < 64) qv[tid] = proj[(size_t)node * PROJ_N + tid];
  if (tid < 48) qp[tid] = pts[(size_t)node * 192 + tid];
  if (tid < 12) Rs[tid] = Rt[(size_t)node * 12 + tid];
  if (tid < KNB) Es[tid] = Eidx[(size_t)node * KNB + tid];
  __syncthreads();

  // gather neighbor k/v (b128, L2-resident): 30 rows x 16 float4
  for (int e = tid; e < KNB * 16; e += 128) {
    const int j = e >> 4;
    const int q4 = e & 15;           // float4 index within the 64-float row
    const int h = q4 >> 2, s4 = q4 & 3;
    const float4* pr = (const float4*)(proj + (size_t)(b * NN + Es[j]) * PROJ_N);
    const float4 kv4 = pr[16 + h * 8 + s4];       // col 64 + h*32 + s4*4
    const float4 vv4 = pr[16 + h * 8 + 4 + s4];   // col 64 + h*32 + 16 + s4*4
    const int c = h * 16 + s4 * 4;
    kbn[j][c] = kv4.x; kbn[j][c+1] = kv4.y; kbn[j][c+2] = kv4.z; kbn[j][c+3] = kv4.w;
    vbn[j][c] = vv4.x; vbn[j][c+1] = vv4.y; vbn[j][c+2] = vv4.z; vbn[j][c+3] = vv4.w;
  }
  // gather k_pts: 30 x 12 float4
  for (int e = tid; e < KNB * 12; e += 128) {
    const int j = e / 12, c4 = e % 12;
    const float4 v = ((const float4*)(pts + (size_t)(b * NN + Es[j]) * 192 + 48))[c4];
    const int c = c4 * 4;
    kpt[j][c] = v.x; kpt[j][c+1] = v.y; kpt[j][c+2] = v.z; kpt[j][c+3] = v.w;
  }
  // gather v_pts: 30 x 24 float4
  for (int e = tid; e < KNB * 24; e += 128) {
    const int j = e / 24, c4 = e % 24;
    const float4 v = ((const float4*)(pts + (size_t)(b * NN + Es[j]) * 192 + 96))[c4];
    const int c = c4 * 4;
    vpt[j][c] = v.x; vpt[j][c+1] = v.y; vpt[j][c+2] = v.z; vpt[j][c+3] = v.w;
  }
  __syncthreads();

  // edge bias + attention logits: one thread per (j,h)
  if (tid < KNB * HH) {
    const int j = tid >> 2, h = tid & 3;
    float bias = bb_g[h];
    for (int e = 0; e < EDIM; ++e) bias += hE[j][e] * Wb[e * HH + h];
    float qk = 0.f;
    for (int c = 0; c < CC; ++c) qk += qv[h * CC + c] * kbn[j][h * CC + c];
    float pa = 0.f;
    for (int p = 0; p < PQN; ++p) {
      const int mb = (h * PQN + p) * 3;
      const float dx = qp[mb] - kpt[j][mb];
      const float dy = qp[mb + 1] - kpt[j][mb + 1];
      const float dz = qp[mb + 2] - kpt[j][mb + 2];
      pa += dx * dx + dy * dy + dz * dz;
    }
    const float x = hwv[h];
    const float hw = (x > 20.f) ? x : log1pf(expf(x));   // softplus
    lg[j][h] = qk * 0.14433756729740643f                 // sqrt(1/(3C))
             + 0.5773502691896258f * bias                // sqrt(1/3)
             - 0.5f * 0.13608276348795434f * hw * pa     // sqrt(1/54)
             + 100000.0f * (mask[(size_t)node * KNB + j] - 1.0f);
  }
  __syncthreads();

  // softmax over K per head
  if (tid < HH) {
    const int h = tid;
    float m = -3.4e38f;
    for (int j = 0; j < KNB; ++j) m = fmaxf(m, lg[j][h]);
    float s = 0.f;
    for (int j = 0; j < KNB; ++j) s += expf(lg[j][h] - m);
    const float inv = 1.0f / s;
    for (int j = 0; j < KNB; ++j) apad[h][j] = expf(lg[j][h] - m) * inv;
  }
  __syncthreads();

  // o (64) and o_pt (96) weighted sums
  for (int e = tid; e < 160; e += 128) {
    if (e < 64) {
      const int h = e >> 4;
      float s = 0.f;
      for (int j = 0; j < KNB; ++j) s += apad[h][j] * vbn[j][e];
      osum[e] = s;
    } else {
      const int idx = e - 64;
      const int h = idx / 24;   // 8 points * 3 coords per head
      float s = 0.f;
      for (int j = 0; j < KNB; ++j) s += apad[h][j] * vpt[j][idx];
      optv[idx] = s;
    }
  }
  __syncthreads();

  float* so = s_in + (size_t)node * OUT_IN;
  if (tid < 64) so[tid] = osum[tid];
  if (tid < 32) {
    const int m = tid;
    const float dx = optv[m*3 + 0] - Rs[9];
    const float dy = optv[m*3 + 1] - Rs[10];
    const float dz = optv[m*3 + 2] - Rs[11];
    // inverse rotation: out[i] = sum_j R[j][i] * d[j]
    const float ox = Rs[0]*dx + Rs[3]*dy + Rs[6]*dz;
    const float oy = Rs[1]*dx + Rs[4]*dy + Rs[7]*dz;
    const float oz = Rs[2]*dx + Rs[5]*dy + Rs[8]*dz;
    so[64 + m] = ox; so[96 + m] = oy; so[128 + m] = oz;
    so[160 + m] = sqrtf(ox*ox + oy*oy + oz*oz + 1e-8f);
  }

  // o_pair = a (4x30, padded 16x32) @ h_E (30x128, padded 32x128) via WMMA
  const int wave = tid >> 5, lane = tid & 31;
  const int lrow = lane & 15;
  const int koff = (lane < 16) ? 0 : 2;
  for (int tt = 0; tt < 2; ++tt) {
    const int ntile = wave * 2 + tt;
    const int n = ntile * 16 + lrow;
    v8f acc = {0.f, 0.f, 0.f, 0.f, 0.f, 0.f, 0.f, 0.f};
    for (int k0 = 0; k0 < 32; k0 += 4) {
      v2f av; av[0] = apad[lrow][k0 + koff]; av[1] = apad[lrow][k0 + koff + 1];
      v2f bv; bv[0] = hE[k0 + koff][n];      bv[1] = hE[k0 + koff + 1][n];
      acc = __builtin_amdgcn_wmma_f32_16x16x4_f32(false, av, false, bv, (short)0, acc, false, false);
    }
    if (lane < 16) {   // only rows M=0..3 (heads) are valid
#pragma unroll
      for (int r = 0; r < 4; ++r) so[192 + r * 128 + n] = acc[r];
    }
  }
}

// ---------------------------------------------------------------------------
// Kernel D: final GEMM (16384 x 704) @ (704 x 128) + bias -> d_out
// ---------------------------------------------------------------------------
__global__ void __launch_bounds__(128)
out_gemm(const float* __restrict__ s_in, const float* __restrict__ Wout,
         const float* __restrict__ bout, float* __restrict__ out) {
  const int wave = threadIdx.x >> 5, lane = threadIdx.x & 31;
  const int row0 = blockIdx.x * 16;
  const int ntile = blockIdx.y * 4 + wave;       // 0..7
  const int n = ntile * 16 + (lane & 15);
  const int mrow = row0 + (lane & 15);
  const int koff = (lane < 16) ? 0 : 2;
  const float* arow = s_in + (size_t)mrow * OUT_IN;
  v8f acc = {0.f, 0.f, 0.f, 0.f, 0.f, 0.f, 0.f, 0.f};
  for (int k0 = 0; k0 < OUT_IN; k0 += 4) {
    v2f av; av[0] = arow[k0 + koff];              av[1] = arow[k0 + koff + 1];
    v2f bv; bv[0] = Wout[(k0 + koff) * DD + n];   bv[1] = Wout[(k0 + koff + 1) * DD + n];
    acc = __builtin_amdgcn_wmma_f32_16x16x4_f32(false, av, false, bv, (short)0, acc, false, false);
  }
  const float bias = bout[n];
  const int mbase = row0 + ((lane < 16) ? 0 : 8);
#pragma unroll
  for (int r = 0; r < 8; ++r)
    out[(size_t)(mbase + r) * DD + n] = acc[r] + bias;
}

// ---------------------------------------------------------------------------
extern "C" void kernel_launch(void* const* d_in, const int* in_sizes, int n_in,
                              void* d_out, int out_size, void* d_ws, size_t ws_size,
                              hipStream_t stream) {
  const float* hV   = (const float*)d_in[0];
  const float* hEg  = (const float*)d_in[1];
  const float* X    = (const float*)d_in[2];
  const int*   Eidx = (const int*)  d_in[3];
  const float* mask = (const float*)d_in[4];
  const float* Wq   = (const float*)d_in[5];
  const float* bq   = (const float*)d_in[6];
  const float* Wkv  = (const float*)d_in[7];
  const float* bkv  = (const float*)d_in[8];
  const float* Wqp  = (const float*)d_in[9];
  const float* bqp  = (const float*)d_in[10];
  const float* Wkvp = (const float*)d_in[11];
  const float* bkvp = (const float*)d_in[12];
  const float* Wb   = (const float*)d_in[13];
  const float* bb   = (const float*)d_in[14];
  const float* hw   = (const float*)d_in[15];
  const float* Wout = (const float*)d_in[16];
  const float* bout = (const float*)d_in[17];

  float* ws   = (float*)d_ws;
  float* proj = ws;                                  // NODES * 384
  float* Rt   = proj + (size_t)NODES * PROJ_N;       // NODES * 12
  float* pts  = Rt   + (size_t)NODES * 12;           // NODES * 192
  float* sin_ = pts  + (size_t)NODES * 192;          // NODES * 704

  proj_gemm<<<dim3(NODES / 16, 6), dim3(128), 0, stream>>>(
      hV, Wq, bq, Wkv, bkv, Wqp, bqp, Wkvp, bkvp, proj);
  frames_rotate<<<dim3(NODES / 128), dim3(128), 0, stream>>>(X, proj, Rt, pts);
  attn_kernel<<<dim3(NODES), dim3(128), 0, stream>>>(
      hEg, Eidx, mask, Wb, bb, hw, proj, Rt, pts, sin_);
  out_gemm<<<dim3(NODES / 16, 2), dim3(128), 0, stream>>>(sin_, Wout, bout, (float*)d_out);
}